// DynamicsNet_64149631533196
// MI455X (gfx1250) — compile-verified
//
#include <hip/hip_runtime.h>

typedef __attribute__((ext_vector_type(2))) float v2f;
typedef __attribute__((ext_vector_type(8))) float v8f;

#define LDS_STRIDE 50   // padded row stride (floats) -> conflict-free gate tile
#define ROWS 16
#define WPB 8           // waves (tiles) per 256-thread block

#define LDS_FENCE() asm volatile("s_wait_dscnt 0" ::: "memory")

static __device__ __forceinline__ float rcp_f(float x) {
    return __builtin_amdgcn_rcpf(x);
}
static __device__ __forceinline__ float tanh_f(float x) {
#if __has_builtin(__builtin_amdgcn_tanhf)
    return __builtin_amdgcn_tanhf(x);     // CDNA5 V_TANH_F32 (single trans op)
#else
    float e = __expf(-2.0f * x);
    return (1.0f - e) * rcp_f(1.0f + e);
#endif
}
static __device__ __forceinline__ float sigmoid_f(float x) {
    // sigmoid(x) = 0.5 + 0.5*tanh(x/2): 1 trans + mul + fma
    return fmaf(0.5f, tanh_f(0.5f * x), 0.5f);
}
static __device__ __forceinline__ float selu_f(float x) {
    const float a = 1.6732632423543772f, s = 1.0507009873554805f;
    return x > 0.0f ? s * x : s * a * (__expf(x) - 1.0f);
}
// LSTM cell with h0=c0=0: c = sig(i)*tanh(g); h = sig(o)*tanh(c). (f gate drops out.)
static __device__ __forceinline__ float lstm_h(float gi, float gg, float go) {
    float c = sigmoid_f(gi) * tanh_f(gg);
    return sigmoid_f(go) * tanh_f(c);
}
static __device__ __forceinline__ v8f splat8(float v) {
    v8f r = {v, v, v, v, v, v, v, v};
    return r;
}
static __device__ __forceinline__ v8f wmma4(v2f a, v2f b, v8f c) {
    // D = A(16x4 f32) x B(4x16 f32) + C(16x16 f32)
    return __builtin_amdgcn_wmma_f32_16x16x4_f32(false, a, false, b, (short)0, c,
                                                 false, false);
}

// Store three 16x16 C tiles (gates, bias already accumulated) into LDS [16][48].
static __device__ __forceinline__ void store_gates(float* g, const v8f* acc,
                                                   int half, int l16) {
#pragma unroll
    for (int t = 0; t < 3; ++t)
#pragma unroll
        for (int r = 0; r < 8; ++r)
            g[(r + half * 8) * LDS_STRIDE + t * 16 + l16] = acc[t][r];
}

// Each lane produces the 6 raw LSTM-h values its next-layer A fragments need:
// A slot (step s, vgpr v) = h[m = lane%16, k = 4s + 2*half + v].
static __device__ __forceinline__ void cell_slots(const float* g, int half, int l16,
                                                  float* hr) {
#pragma unroll
    for (int s = 0; s < 3; ++s)
#pragma unroll
        for (int v = 0; v < 2; ++v) {
            const int k = s * 4 + half * 2 + v;
            const float* row = g + l16 * LDS_STRIDE;
            hr[s * 2 + v] = lstm_h(row[k], row[24 + k], row[36 + k]);
        }
}

// tanh(cell) outputs, lane-linear so global stores coalesce to 128B/instr.
static __device__ __forceinline__ void emit_out(const float* g, float* dst, int lane) {
#pragma unroll
    for (int q = 0; q < 6; ++q) {
        const int flat = q * 32 + lane;        // 0..191 = row*12 + j
        const int r = flat / 12;
        const int j = flat - r * 12;
        const float* row = g + r * LDS_STRIDE;
        dst[flat] = tanh_f(lstm_h(row[j], row[24 + j], row[36 + j]));
    }
}

__global__ __launch_bounds__(256) void dynamics_net_kernel(
    const float* __restrict__ x,
    const float* __restrict__ W1, const float* __restrict__ bi1, const float* __restrict__ bh1,
    const float* __restrict__ W2, const float* __restrict__ bi2, const float* __restrict__ bh2,
    const float* __restrict__ Wm, const float* __restrict__ bim, const float* __restrict__ bhm,
    const float* __restrict__ Wd, const float* __restrict__ bid, const float* __restrict__ bhd,
    float* __restrict__ out, int n_tiles, long long rowsB) {
    __shared__ float lds[WPB * ROWS * LDS_STRIDE];

    const int lane = threadIdx.x & 31;
    const int wave = threadIdx.x >> 5;
    const int half = lane >> 4;   // 0: lanes 0-15, 1: lanes 16-31
    const int l16  = lane & 15;
    const int tile = blockIdx.x * WPB + wave;
    if (tile >= n_tiles) return;  // wave-uniform: EXEC stays all-ones for WMMA

    float* g = lds + wave * (ROWS * LDS_STRIDE);

    // ---- Preload weight B fragments: B[k,n] = W[n,k]; lane supplies (n=16t+l16,
    // k=4s+2*half+{0,1}) as one float2. W1:[48,16] K=16 (4 steps); others:[48,12] (3). ----
    v2f bw1[3][4];
#pragma unroll
    for (int t = 0; t < 3; ++t) {
        const float* wr = W1 + (t * 16 + l16) * 16 + half * 2;
#pragma unroll
        for (int s = 0; s < 4; ++s) bw1[t][s] = *(const v2f*)(wr + s * 4);
    }
    v2f bw2[3][3], bwm[3][3], bwd[3][3];
#pragma unroll
    for (int t = 0; t < 3; ++t) {
        const int n = t * 16 + l16;
#pragma unroll
        for (int s = 0; s < 3; ++s) {
            bw2[t][s] = *(const v2f*)(W2 + n * 12 + half * 2 + s * 4);
            bwm[t][s] = *(const v2f*)(Wm + n * 12 + half * 2 + s * 4);
            bwd[t][s] = *(const v2f*)(Wd + n * 12 + half * 2 + s * 4);
        }
    }
    float bs1[3], bs2[3], bsm[3], bsd[3];
#pragma unroll
    for (int t = 0; t < 3; ++t) {
        const int n = t * 16 + l16;
        bs1[t] = bi1[n] + bh1[n];
        bs2[t] = bi2[n] + bh2[n];
        bsm[t] = bim[n] + bhm[n];
        bsd[t] = bid[n] + bhd[n];
    }

    // ---- x tile as A fragments (16 rows x K=16, exactly once per element) ----
    const float* xr = x + ((long long)tile * 16 + l16) * 16 + half * 2;
    v2f xa[4];
#pragma unroll
    for (int s = 0; s < 4; ++s) xa[s] = *(const v2f*)(xr + s * 4);

    v8f acc[3];
    float hr[6];
    v2f ha[3];

    // ======== layer 1: gates = x @ W1^T + b  (C init = bias splat) ========
#pragma unroll
    for (int t = 0; t < 3; ++t) {
        v8f c = splat8(bs1[t]);
#pragma unroll
        for (int s = 0; s < 4; ++s) c = wmma4(xa[s], bw1[t][s], c);
        acc[t] = c;
    }
    store_gates(g, acc, half, l16);
    LDS_FENCE();
    cell_slots(g, half, l16, hr);                       // h1 = relu(cell)
#pragma unroll
    for (int s = 0; s < 3; ++s) {
        v2f a = {fmaxf(hr[2 * s], 0.0f), fmaxf(hr[2 * s + 1], 0.0f)};
        ha[s] = a;
    }
    LDS_FENCE();

    // ======== layer 2: gates = h1 @ W2^T + b ========
#pragma unroll
    for (int t = 0; t < 3; ++t) {
        v8f c = splat8(bs2[t]);
#pragma unroll
        for (int s = 0; s < 3; ++s) c = wmma4(ha[s], bw2[t][s], c);
        acc[t] = c;
    }
    store_gates(g, acc, half, l16);
    LDS_FENCE();
    cell_slots(g, half, l16, hr);                       // h2 = selu(cell)
#pragma unroll
    for (int s = 0; s < 3; ++s) {
        v2f a = {selu_f(hr[2 * s]), selu_f(hr[2 * s + 1])};
        ha[s] = a;
    }
    LDS_FENCE();

    // ======== head m: tanh(cell(h2 @ Wm^T + b)) ========
#pragma unroll
    for (int t = 0; t < 3; ++t) {
        v8f c = splat8(bsm[t]);
#pragma unroll
        for (int s = 0; s < 3; ++s) c = wmma4(ha[s], bwm[t][s], c);
        acc[t] = c;
    }
    store_gates(g, acc, half, l16);
    LDS_FENCE();
    emit_out(g, out + (long long)tile * 192, lane);
    LDS_FENCE();

    // ======== head d: tanh(cell(h2 @ Wd^T + b)) ========
#pragma unroll
    for (int t = 0; t < 3; ++t) {
        v8f c = splat8(bsd[t]);
#pragma unroll
        for (int s = 0; s < 3; ++s) c = wmma4(ha[s], bwd[t][s], c);
        acc[t] = c;
    }
    store_gates(g, acc, half, l16);
    LDS_FENCE();
    emit_out(g, out + rowsB * 12 + (long long)tile * 192, lane);
}

extern "C" void kernel_launch(void* const* d_in, const int* in_sizes, int n_in,
                              void* d_out, int out_size, void* d_ws, size_t ws_size,
                              hipStream_t stream) {
    const float* x   = (const float*)d_in[0];
    const float* W1  = (const float*)d_in[1];
    const float* bi1 = (const float*)d_in[2];
    const float* bh1 = (const float*)d_in[3];
    const float* W2  = (const float*)d_in[4];
    const float* bi2 = (const float*)d_in[5];
    const float* bh2 = (const float*)d_in[6];
    const float* Wm  = (const float*)d_in[7];
    const float* bim = (const float*)d_in[8];
    const float* bhm = (const float*)d_in[9];
    const float* Wd  = (const float*)d_in[10];
    const float* bid = (const float*)d_in[11];
    const float* bhd = (const float*)d_in[12];
    float* out = (float*)d_out;

    const long long rowsB = (long long)in_sizes[0] / 16;   // B (IN = 16)
    const int n_tiles = (int)((rowsB + 15) / 16);
    const int blocks  = (n_tiles + WPB - 1) / WPB;

    hipLaunchKernelGGL(dynamics_net_kernel, dim3(blocks), dim3(256), 0, stream,
                       x, W1, bi1, bh1, W2, bi2, bh2, Wm, bim, bhm, Wd, bid, bhd,
                       out, n_tiles, rowsB);
}